// MLP9HiddenLayers_7911329759901
// MI455X (gfx1250) — compile-verified
//
#include <hip/hip_runtime.h>

typedef __attribute__((ext_vector_type(2))) float v2f;
typedef __attribute__((ext_vector_type(8))) float v8f;

#define NEG_SLOPE 0.01f
#define HID 7
#define N_MID 8
#define TPI 4                 // 16-row tiles per wave iteration (64 rows)
#define NSLOT (1 + 2*N_MID + 2)   // 19 A-operand slots: A0, Am[8][2], A9[2]
#define WS_BYTES (NSLOT * 32 * sizeof(float2))

__device__ __forceinline__ float lk(float v) {
    // leaky_relu, slope in (0,1): max(x, slope*x)
    return fmaxf(v, NEG_SLOPE * v);
}

// Swap lanes 0-15 <-> 16-31 (wave32): ds_swizzle group-of-32, xor=0x10, and=0x1f.
__device__ __forceinline__ float swap16(float v) {
    return __builtin_bit_cast(float,
        __builtin_amdgcn_ds_swizzle(__builtin_bit_cast(int, v), 0x401f));
}

__device__ __forceinline__ v8f wmma4(v2f a, v2f b, v8f c) {
    // D(16x16,f32) = A(16x4,f32) x B(4x16,f32) + C
    return __builtin_amdgcn_wmma_f32_16x16x4_f32(false, a, false, b, (short)0, c, false, false);
}

struct WSet { v2f A0; v2f Am[N_MID][2]; v2f A9[2]; };

// Build this lane's slice of all 10 layers' A operands (feature->row permutation:
// rows {0,1,2,3}<-feats{0,1,4,5}, rows {8,9,10}<-feats{2,3,6}; K == input feature),
// so the next layer's B operand is exactly D v0..v3 with no data movement.
__device__ __forceinline__ void buildW(int lane, const float* __restrict__ W0,
                                       const float* __restrict__ Wm,
                                       const float* __restrict__ W9, WSet& w) {
    const int  r   = lane & 15;
    const bool hi  = (lane & 16) != 0;
    const bool lo4 = (r < 4);
    const bool hig = (r >= 8) && (r <= 10);
    const bool rok = lo4 || hig;
    const int  f   = lo4 ? ((r < 2) ? r : r + 2) : ((r == 10) ? 6 : r - 6);
    const int  fc  = rok ? f : 0;

    const bool lov = rok && !hi;              // layer 0: K=2,3 columns are zero (IN_DIM=2)
    w.A0[0] = lov ? W0[fc*2+0] : 0.f;
    w.A0[1] = lov ? W0[fc*2+1] : 0.f;
#pragma unroll
    for (int i = 0; i < N_MID; ++i) {
        const float* W = Wm + i*HID*HID + fc*HID;      // permuted row of W[i]
        w.Am[i][0][0] = rok ? (hi ? W[2] : W[0]) : 0.f;  // K0 | K2
        w.Am[i][0][1] = rok ? (hi ? W[3] : W[1]) : 0.f;  // K1 | K3
        w.Am[i][1][0] = rok ? (hi ? W[6] : W[4]) : 0.f;  // K4 | K6
        w.Am[i][1][1] = (rok && !hi) ? W[5] : 0.f;       // K5 | K7(pad)
    }
    const bool r0 = (r == 0);                 // final layer: single output row 0
    w.A9[0][0] = r0 ? (hi ? W9[2] : W9[0]) : 0.f;
    w.A9[0][1] = r0 ? (hi ? W9[3] : W9[1]) : 0.f;
    w.A9[1][0] = r0 ? (hi ? W9[6] : W9[4]) : 0.f;
    w.A9[1][1] = (r0 && !hi) ? W9[5] : 0.f;
}

// One wave: gather/permute weights once, store lane-major so the main kernel
// reads them with 19 coalesced unpredicated b64 loads.
__global__ void __launch_bounds__(32)
mlp10_prep_kernel(const float* __restrict__ W0, const float* __restrict__ Wm,
                  const float* __restrict__ W9, float2* __restrict__ ws) {
    const int lane = threadIdx.x & 31;
    WSet w;
    buildW(lane, W0, Wm, W9, w);
    ws[ 0*32 + lane] = make_float2(w.A0[0], w.A0[1]);
#pragma unroll
    for (int i = 0; i < N_MID; ++i) {
        ws[(1 + 2*i    )*32 + lane] = make_float2(w.Am[i][0][0], w.Am[i][0][1]);
        ws[(1 + 2*i + 1)*32 + lane] = make_float2(w.Am[i][1][0], w.Am[i][1][1]);
    }
    ws[(1 + 2*N_MID    )*32 + lane] = make_float2(w.A9[0][0], w.A9[0][1]);
    ws[(1 + 2*N_MID + 1)*32 + lane] = make_float2(w.A9[1][0], w.A9[1][1]);
}

template <bool FROMWS>
__global__ void __launch_bounds__(256)
mlp10_wmma_kernel(const float* __restrict__ x,
                  const float* __restrict__ W0,
                  const float* __restrict__ Wm,
                  const float* __restrict__ W9,
                  const float2* __restrict__ ws,
                  float* __restrict__ out, int batch)
{
    const int  lane = threadIdx.x & 31;
    const bool hi   = (lane & 16) != 0;

    WSet w;
    if (FROMWS) {
        float2 t;
        t = ws[0*32 + lane];               w.A0[0] = t.x;       w.A0[1] = t.y;
#pragma unroll
        for (int i = 0; i < N_MID; ++i) {
            t = ws[(1 + 2*i    )*32 + lane]; w.Am[i][0][0] = t.x; w.Am[i][0][1] = t.y;
            t = ws[(1 + 2*i + 1)*32 + lane]; w.Am[i][1][0] = t.x; w.Am[i][1][1] = t.y;
        }
        t = ws[(1 + 2*N_MID    )*32 + lane]; w.A9[0][0] = t.x;   w.A9[0][1] = t.y;
        t = ws[(1 + 2*N_MID + 1)*32 + lane]; w.A9[1][0] = t.x;   w.A9[1][1] = t.y;
    } else {
        buildW(lane, W0, Wm, W9, w);
    }

    const int ntiles  = (batch + 15) >> 4;
    const int ngroups = (ntiles + TPI - 1) / TPI;
    const int wid     = blockIdx.x * (blockDim.x >> 5) + (threadIdx.x >> 5);
    const int nw      = gridDim.x * (blockDim.x >> 5);
    const float2* x2  = (const float2*)x;
    const int maxIdx  = batch - 1;

    if (wid >= ngroups) return;            // uniform per wave

    // Full-width clamped loads: lanes 0-31 cover 32 consecutive batch rows.
    auto loadpair = [&](int p, float2& a, float2& b) {
        int base = p * (TPI * 16);
        int i0 = base + lane;       i0 = (i0 > maxIdx) ? maxIdx : i0;
        int i1 = base + 32 + lane;  i1 = (i1 > maxIdx) ? maxIdx : i1;
        a = x2[i0];
        b = x2[i1];
    };

    float2 cx0, cx1;
    loadpair(wid, cx0, cx1);

    for (int p = wid; p < ngroups; p += nw) {
        float2 nx0, nx1;
        loadpair(p + nw, nx0, nx1);        // prefetch next group (clamped, always safe)

        v8f d[TPI];
        // Layer 0: B V0 = x[:,0]|dontcare, V1 = x[:,1]|dontcare (A0 K2/K3 cols are 0).
#pragma unroll
        for (int tt = 0; tt < TPI; ++tt) {
            float2 src = (tt < 2) ? cx0 : cx1;
            v2f b;
            if (tt & 1) { b[0] = swap16(src.x); b[1] = swap16(src.y); }
            else        { b[0] = src.x;         b[1] = src.y;         }
            v8f c = {};
            d[tt] = wmma4(w.A0, b, c);     // D rows permuted; unused rows stay 0
        }

        // Hidden layers: B chunk0 = {lk(d0),lk(d1)}, chunk1 = {lk(d2),lk(d3)}.
#pragma unroll
        for (int L = 0; L < N_MID; ++L) {
#pragma unroll
            for (int tt = 0; tt < TPI; ++tt) {
                v2f b0, b1;
                b0[0] = lk(d[tt][0]); b0[1] = lk(d[tt][1]);
                b1[0] = lk(d[tt][2]); b1[1] = lk(d[tt][3]);
                v8f c = {};
                c     = wmma4(w.Am[L][0], b0, c);
                d[tt] = wmma4(w.Am[L][1], b1, c);
            }
        }

        // Output layer (7->1) + final leaky; result = D row 0 (lanes 0-15 of v0).
        float res[TPI];
#pragma unroll
        for (int tt = 0; tt < TPI; ++tt) {
            v2f b0, b1;
            b0[0] = lk(d[tt][0]); b0[1] = lk(d[tt][1]);
            b1[0] = lk(d[tt][2]); b1[1] = lk(d[tt][3]);
            v8f c = {};
            c     = wmma4(w.A9[0], b0, c);
            v8f o = wmma4(w.A9[1], b1, c);
            res[tt] = lk(o[0]);
        }

        // Pack tile pairs into full-wave 32-lane stores (one swizzle per pair).
        const int base = p * (TPI * 16);
#pragma unroll
        for (int g = 0; g < TPI / 2; ++g) {
            float v  = hi ? swap16(res[2*g + 1]) : res[2*g];
            int  row = base + g*32 + lane;
            if (row < batch) out[row] = v;
        }

        cx0 = nx0; cx1 = nx1;
    }
}

extern "C" void kernel_launch(void* const* d_in, const int* in_sizes, int n_in,
                              void* d_out, int out_size, void* d_ws, size_t ws_size,
                              hipStream_t stream) {
    const float* x  = (const float*)d_in[0];   // [B,2]
    const float* W0 = (const float*)d_in[1];   // [7,2]
    const float* Wm = (const float*)d_in[2];   // [8,7,7]
    const float* W9 = (const float*)d_in[3];   // [1,7]
    float* out = (float*)d_out;                // [B,1]
    const int batch = in_sizes[0] / 2;

    const int threads = 256;                   // 8 wave32 per block
    const int ntiles  = (batch + 15) >> 4;
    const int ngroups = (ntiles + TPI - 1) / TPI;
    const int wavesPerBlock = threads >> 5;
    int blocks = 2048;
    const int maxBlocks = (ngroups + wavesPerBlock - 1) / wavesPerBlock;
    if (blocks > maxBlocks) blocks = (maxBlocks > 0) ? maxBlocks : 1;

    if (ws_size >= WS_BYTES) {
        float2* ws = (float2*)d_ws;
        mlp10_prep_kernel<<<1, 32, 0, stream>>>(W0, Wm, W9, ws);
        mlp10_wmma_kernel<true><<<blocks, threads, 0, stream>>>(x, W0, Wm, W9, ws, out, batch);
    } else {
        mlp10_wmma_kernel<false><<<blocks, threads, 0, stream>>>(x, W0, Wm, W9, nullptr, out, batch);
    }
}